// InstanceDecoder_35493609734731
// MI455X (gfx1250) — compile-verified
//
#include <hip/hip_runtime.h>
#include <math.h>

typedef float v2f __attribute__((ext_vector_type(2)));
typedef float v8f __attribute__((ext_vector_type(8)));
typedef int   v4i __attribute__((ext_vector_type(4)));
typedef int   v8i __attribute__((ext_vector_type(8)));

#define N_HITS 49152
#define HIT_F  256
#define INST_F 32
#define K_INST 1024
#define QMINF  0.5f
#define SBF    1.0f
#define XSTR   36   // padded LDS stride (floats) for 32-wide xa rows (TDM pad produces this)
#define XENC   20   // padded LDS stride (floats) for 16-wide x chunks

// scal[] slots: 0=v_att  1=rep_total  2=rep_member  3=sum(1-beta_a)  4=noise_sum
__global__ void k_init(unsigned* __restrict__ segbits, int* __restrict__ alpha,
                       float* __restrict__ scal, int* __restrict__ cnt) {
  int t = blockIdx.x * blockDim.x + threadIdx.x;
  if (t < K_INST) { segbits[t] = 0u; alpha[t] = N_HITS; }
  if (t < 8) scal[t] = 0.0f;
  if (t == 0) *cnt = 0;
}

// ox = x@Wc + bc ; beta = sigmoid(x@Wb + bb); q = atanh(clip(beta))^2 + qmin; oxn2 = ||ox||^2
// x tile staged via double-buffered GLOBAL_LOAD_ASYNC_TO_LDS_B128 (ASYNCcnt pipeline).
__global__ __launch_bounds__(256) void k_encode(
    const float* __restrict__ x, const float* __restrict__ Wb, const float* __restrict__ bb,
    const float* __restrict__ Wc, const float* __restrict__ bc,
    float* __restrict__ ox, float* __restrict__ qv, float* __restrict__ betac,
    float* __restrict__ oxn2)
{
  __shared__ float sWc2[HIT_F * INST_F];   // K-pair interleaved: [(k/2)*64 + col*2 + (k&1)]
  __shared__ float sWb[HIT_F];
  __shared__ float sBc[INST_F];
  __shared__ float sX[2][128 * XENC];      // double-buffered 128x16 x tiles
  const int tid = threadIdx.x;
  for (int i = tid; i < HIT_F * INST_F; i += 256) {
    int k = i >> 5, col = i & 31;
    sWc2[(k >> 1) * 64 + col * 2 + (k & 1)] = Wc[i];
  }
  sWb[tid] = Wb[tid];
  if (tid < INST_F) sBc[tid] = bc[tid];

  const int wave = tid >> 5, lane = tid & 31;
  const int half = lane >> 4, l16 = lane & 15;
  const int koff = half * 2;
  const long blockBase = (long)blockIdx.x * 128;
  const int arow = wave * 16 + l16;

  // per-lane async DMA of one 128x16 chunk straight into LDS (2 x b128 per lane)
  auto issue_chunk = [&](int c, float* buf) {
    #pragma unroll
    for (int i = 0; i < 2; ++i) {
      int fl = (tid + i * 256) * 4;
      int r = fl >> 4, cc = fl & 15;
      unsigned ldsa = (unsigned)(unsigned long long)&buf[r * XENC + cc];
      unsigned voff = (unsigned)(((blockBase + r) * HIT_F + c * 16 + cc) * 4);
      asm volatile("global_load_async_to_lds_b128 %0, %1, %2 offset:0"
                   :: "v"(ldsa), "v"(voff), "s"(x) : "memory");
    }
  };

  v8f acc0 = {}; v8f acc1 = {};
  float bp = 0.0f;

  issue_chunk(0, sX[0]);
  issue_chunk(1, sX[1]);

  for (int c = 0; c < 16; ++c) {          // K=256 in chunks of 16, double-buffered
    if (c < 15) asm volatile("s_wait_asynccnt 0x2" ::: "memory");
    else        asm volatile("s_wait_asynccnt 0x0" ::: "memory");
    __syncthreads();                       // chunk c visible to all waves (also sWc2 on c==0)
    const float* sx = sX[c & 1];
    #pragma unroll
    for (int kk = 0; kk < 16; kk += 4) {
      v2f a = *(const v2f*)&sx[arow * XENC + kk + koff];   // A frag: M=l16, K={koff,koff+1}
      int krow = c * 16 + kk + koff;
      bp = fmaf(a.x, sWb[krow], bp);                       // beta GEMV piggyback
      bp = fmaf(a.y, sWb[krow + 1], bp);
      v2f b0 = *(const v2f*)&sWc2[(krow >> 1) * 64 + l16 * 2];        // contiguous b64 frags
      v2f b1 = *(const v2f*)&sWc2[(krow >> 1) * 64 + 32 + l16 * 2];
      acc0 = __builtin_amdgcn_wmma_f32_16x16x4_f32(false, a, false, b0, (short)0, acc0, false, false);
      acc1 = __builtin_amdgcn_wmma_f32_16x16x4_f32(false, a, false, b1, (short)0, acc1, false, false);
    }
    __syncthreads();                       // all waves done with buf[c&1]
    if (c + 2 < 16) issue_chunk(c + 2, sX[c & 1]);
  }

  const long rowBase = blockBase + wave * 16;
  #pragma unroll
  for (int v = 0; v < 8; ++v) {            // C layout: VGPR v -> rows v / v+8 per lane half
    long row = rowBase + v + half * 8;
    float v0 = acc0[v] + sBc[l16];
    float v1 = acc1[v] + sBc[16 + l16];
    ox[row * INST_F + l16]      = v0;
    ox[row * INST_F + 16 + l16] = v1;
    float p = fmaf(v0, v0, v1 * v1);
    #pragma unroll
    for (int m = 1; m < 16; m <<= 1) p += __shfl_xor(p, m);
    if (l16 == 0) oxn2[row] = p;
  }
  bp += __shfl_xor(bp, 16);                // combine K-halves of beta dot
  if (lane < 16) {
    long row = rowBase + lane;
    float be = 1.0f / (1.0f + expf(-(bp + bb[0])));
    be = fminf(fmaxf(be, 1e-4f), 1.0f - 1e-4f);
    float ah = atanhf(be);
    betac[row] = be;
    qv[row] = fmaf(ah, ah, QMINF);
  }
}

__global__ void k_segmax(const int* __restrict__ eidx, int E, const float* __restrict__ qv,
                         unsigned* __restrict__ segbits) {
  int t = blockIdx.x * blockDim.x + threadIdx.x;
  if (t >= E) return;
  int h = eidx[t], i = eidx[E + t];
  atomicMax(&segbits[i], __float_as_uint(qv[h]));   // q > 0 -> uint order == float order
}

__global__ void k_alpha(const int* __restrict__ eidx, int E, const float* __restrict__ qv,
                        const unsigned* __restrict__ segbits, int* __restrict__ alpha) {
  int t = blockIdx.x * blockDim.x + threadIdx.x;
  if (t >= E) return;
  int h = eidx[t], i = eidx[E + t];
  if (__float_as_uint(qv[h]) >= segbits[i]) atomicMin(&alpha[i], h);
}

__global__ void k_gather(const int* __restrict__ alpha, const float* __restrict__ ox,
                         const float* __restrict__ qv, const float* __restrict__ betac,
                         float* __restrict__ xa, float* __restrict__ qa,
                         float* __restrict__ xan2, float* __restrict__ scal) {
  int k = blockIdx.x, d = threadIdx.x;   // 32 threads/block
  int a = alpha[k]; if (a > N_HITS - 1) a = N_HITS - 1;
  float v = ox[(long)a * INST_F + d];
  xa[k * INST_F + d] = v;
  float p = v * v;
  #pragma unroll
  for (int m = 1; m < 32; m <<= 1) p += __shfl_xor(p, m);
  if (d == 0) {
    xan2[k] = p;
    qa[k] = qv[a];
    atomicAdd(&scal[3], 1.0f - betac[a]);
  }
}

// attractive potential + member-pair hinge (subtracted from dense repulsion)
__global__ __launch_bounds__(256) void k_edges(
    const int* __restrict__ eidx, int E, const float* __restrict__ ox,
    const float* __restrict__ xa, const float* __restrict__ qv,
    const float* __restrict__ qa, float* __restrict__ scal) {
  int t = blockIdx.x * blockDim.x + threadIdx.x;
  float att = 0.0f, mem = 0.0f;
  if (t < E) {
    int h = eidx[t], i = eidx[E + t];
    const float* oh = &ox[(long)h * INST_F];
    const float* xi = &xa[i * INST_F];
    float d2 = 0.0f;
    #pragma unroll
    for (int d = 0; d < INST_F; ++d) { float df = oh[d] - xi[d]; d2 = fmaf(df, df, d2); }
    float w = qv[h] * qa[i];
    att = w * d2;
    float dist = sqrtf(fmaxf(d2, 0.0f) + 1e-12f);
    mem = w * fmaxf(1.0f - dist, 0.0f);
  }
  #pragma unroll
  for (int m = 1; m < 32; m <<= 1) { att += __shfl_xor(att, m); mem += __shfl_xor(mem, m); }
  if ((threadIdx.x & 31) == 0) { atomicAdd(&scal[0], att); atomicAdd(&scal[2], mem); }
}

// dense repulsion: WMMA-f32 GEMM ox @ xa^T with fused hinge epilogue.
// xa column super-tiles (128x32 fp32) streamed by the Tensor Data Mover with LDS padding
// (32-DWORD interval + 4-DWORD pad -> 36-DWORD rows), double-buffered on TENSORcnt.
__global__ __launch_bounds__(256) void k_rep(
    const float* __restrict__ ox, const float* __restrict__ oxn2, const float* __restrict__ qv,
    const float* __restrict__ xa, const float* __restrict__ xan2, const float* __restrict__ qa,
    float* __restrict__ scal)
{
  __shared__ float sXa[2][128 * XSTR];
  const int tid = threadIdx.x;
  const int wave = tid >> 5, lane = tid & 31;
  const int half = lane >> 4, l16 = lane & 15;
  const int koff = half * 2;
  const long rowBase = (long)blockIdx.x * 128 + wave * 16;
  const long arow = rowBase + l16;

  // TDM descriptor: 2-D tile (tile_dim0=32, tile_dim1=128) of 4-byte data, pad-enabled.
  auto issue_tile = [&](int st, int buf) {
    unsigned ldsa = (unsigned)(unsigned long long)&sXa[buf][0];
    unsigned long long ga = (unsigned long long)(const void*)(xa + (size_t)st * 128 * INST_F);
    v4i g0; v8i g1;
    g0[0] = 1;                                           // count=1, user descriptor
    g0[1] = (int)ldsa;                                   // lds_addr
    g0[2] = (int)(unsigned)(ga & 0xffffffffu);           // global_addr[31:0]
    g0[3] = (int)((unsigned)((ga >> 32) & 0x1ffffffu) | 0x80000000u); // ga[56:32] | type=2
    g1[0] = (2 << 16) | (1 << 20) | (4 << 22) | (3 << 25); // data_size=4B, pad_en, 32dw int., 4dw pad
    g1[1] = (int)(32u << 16);                            // tensor_dim0 = 32 (bits 79:48)
    g1[2] = (int)(1024u << 16);                          // tensor_dim0 hi | tensor_dim1 lo = 1024
    g1[3] = (int)(32u << 16);                            // tensor_dim1 hi | tile_dim0 = 32
    g1[4] = 128;                                         // tile_dim1 = 128, tile_dim2 = 0
    g1[5] = 32;                                          // tensor_dim0_stride = 32 elements
    g1[6] = 0;
    g1[7] = 0;
    asm volatile("tensor_load_to_lds %0, %1" :: "s"(g0), "s"(g1) : "memory");
  };

  v2f af[8];                             // wave's 16x32 A tile, resident in VGPRs
  #pragma unroll
  for (int s = 0; s < 8; ++s)
    af[s] = *(const v2f*)&ox[arow * INST_F + s * 4 + koff];

  float qrow[8], nrow[8];
  #pragma unroll
  for (int v = 0; v < 8; ++v) {
    long row = rowBase + v + half * 8;
    qrow[v] = qv[row];
    nrow[v] = oxn2[row];
  }

  if (wave == 0) { issue_tile(0, 0); issue_tile(1, 1); }

  float local = 0.0f;
  for (int st = 0; st < 8; ++st) {       // 8 super-tiles of 128 instance columns
    if (wave == 0) {
      if (st < 7) __builtin_amdgcn_s_wait_tensorcnt(1);
      else        __builtin_amdgcn_s_wait_tensorcnt(0);
    }
    __syncthreads();                     // tile st complete + visible to all waves
    const float* sxa = sXa[st & 1];
    for (int ct = 0; ct < 8; ++ct) {
      int colL = ct * 16 + l16;
      v8f acc = {};
      #pragma unroll
      for (int s = 0; s < 8; ++s) {
        v2f b = *(const v2f*)&sxa[colL * XSTR + s * 4 + koff];
        acc = __builtin_amdgcn_wmma_f32_16x16x4_f32(false, af[s], false, b, (short)0, acc, false, false);
      }
      int colG = st * 128 + colL;
      float qc = qa[colG], nc = xan2[colG];
      #pragma unroll
      for (int v = 0; v < 8; ++v) {      // fused epilogue: hinge(1-dist) weighted sum
        float d2 = nrow[v] + nc - 2.0f * acc[v];
        float dist = sqrtf(fmaxf(d2, 0.0f) + 1e-12f);
        float hinge = fmaxf(1.0f - dist, 0.0f);
        local = fmaf(qrow[v] * qc, hinge, local);
      }
    }
    __syncthreads();                     // all waves done reading buf[st&1]
    if (wave == 0 && st + 2 < 8) issue_tile(st + 2, st & 1);
  }
  #pragma unroll
  for (int m = 1; m < 32; m <<= 1) local += __shfl_xor(local, m);
  if (lane == 0) atomicAdd(&scal[1], local);
}

__global__ void k_noise(const long long* __restrict__ yi, const float* __restrict__ betac,
                        float* __restrict__ scal, int* __restrict__ cnt) {
  int t = blockIdx.x * blockDim.x + threadIdx.x;
  if (t >= N_HITS) return;
  if (yi[t] < 0) { atomicAdd(&scal[4], betac[t]); atomicAdd(cnt, 1); }
}

__global__ void k_final(const float* __restrict__ scal, const int* __restrict__ cnt,
                        const float* __restrict__ temp, float* __restrict__ out) {
  float l_v = (scal[0] + (scal[1] - scal[2])) / (float)N_HITS;
  int nn = *cnt; if (nn < 1) nn = 1;
  float l_b = scal[3] / (float)K_INST + SBF * scal[4] / (float)nn;
  float t = temp[0];
  out[0] = (l_b + l_v) * expf(-t) + t;
}

extern "C" void kernel_launch(void* const* d_in, const int* in_sizes, int n_in,
                              void* d_out, int out_size, void* d_ws, size_t ws_size,
                              hipStream_t stream) {
  (void)n_in; (void)out_size; (void)ws_size;
  const float*     x    = (const float*)d_in[0];
  const float*     Wb   = (const float*)d_in[1];
  const float*     bb   = (const float*)d_in[2];
  const float*     Wc   = (const float*)d_in[3];
  const float*     bc   = (const float*)d_in[4];
  const float*     temp = (const float*)d_in[5];
  const long long* yi   = (const long long*)d_in[6];   // int64 labels
  const int*       eidx = (const int*)d_in[8];         // [2,E] row-major
  const int E = in_sizes[8] / 2;
  float* out = (float*)d_out;

  char* w = (char*)d_ws;
  float* ox      = (float*)w;  w += sizeof(float) * (size_t)N_HITS * INST_F;
  float* qv      = (float*)w;  w += sizeof(float) * N_HITS;
  float* betac   = (float*)w;  w += sizeof(float) * N_HITS;
  float* oxn2    = (float*)w;  w += sizeof(float) * N_HITS;
  float* xa      = (float*)w;  w += sizeof(float) * K_INST * INST_F;
  float* qa      = (float*)w;  w += sizeof(float) * K_INST;
  float* xan2    = (float*)w;  w += sizeof(float) * K_INST;
  unsigned* segb = (unsigned*)w; w += sizeof(unsigned) * K_INST;
  int* alpha     = (int*)w;    w += sizeof(int) * K_INST;
  float* scal    = (float*)w;  w += sizeof(float) * 8;
  int* cnt       = (int*)w;

  k_init  <<<(K_INST + 255) / 256, 256, 0, stream>>>(segb, alpha, scal, cnt);
  k_encode<<<N_HITS / 128, 256, 0, stream>>>(x, Wb, bb, Wc, bc, ox, qv, betac, oxn2);
  k_segmax<<<(E + 255) / 256, 256, 0, stream>>>(eidx, E, qv, segb);
  k_alpha <<<(E + 255) / 256, 256, 0, stream>>>(eidx, E, qv, segb, alpha);
  k_gather<<<K_INST, 32, 0, stream>>>(alpha, ox, qv, betac, xa, qa, xan2, scal);
  k_edges <<<(E + 255) / 256, 256, 0, stream>>>(eidx, E, ox, xa, qv, qa, scal);
  k_rep   <<<N_HITS / 128, 256, 0, stream>>>(ox, oxn2, qv, xa, xan2, qa, scal);
  k_noise <<<(N_HITS + 255) / 256, 256, 0, stream>>>(yi, betac, scal, cnt);
  k_final <<<1, 1, 0, stream>>>(scal, cnt, temp, out);
}